// DecoderLstm_64974265253988
// MI455X (gfx1250) — compile-verified
//
#include <hip/hip_runtime.h>
#include <hip/hip_bf16.h>

typedef float v2f __attribute__((ext_vector_type(2)));
typedef float v8f __attribute__((ext_vector_type(8)));

#define HID   512
#define BATCH 16
#define SEQ   128
#define VOCAB 32000
#define G4    2048   // 4*HID

__device__ __forceinline__ v8f wmma_f32(v2f a, v2f b, v8f c) {
    // D = A(16x4) * B(4x16) + C(16x16), all fp32
    return __builtin_amdgcn_wmma_f32_16x16x4_f32(false, a, false, b, (short)0, c, false, false);
}

__device__ __forceinline__ float sigm(float x) { return 1.0f / (1.0f + __expf(-x)); }

// ---------------------------------------------------------------------------
// init h0/c0 from z: z is (1,B,2H) -> h0=z[...,:H], c0=z[...,H:]
__global__ void init_hc_kernel(const float* __restrict__ z,
                               float* __restrict__ h0, float* __restrict__ c0) {
    int t = blockIdx.x * 256 + threadIdx.x;           // 0 .. B*H-1
    if (t >= BATCH * HID) return;
    int b = t / HID, j = t % HID;
    h0[t] = z[b * (2 * HID) + j];
    c0[t] = z[b * (2 * HID) + HID + j];
}

// ---------------------------------------------------------------------------
// xg[s][b][g] = sum_e emb[x[b][s]][e] * W_ih[g][e] + b_ih[g] + b_hh[g]
// M = S*B (row m = s*16+b), N = 4H, K = 512.  One wave per 16x16 tile.
__global__ void xg_gemm_kernel(const int* __restrict__ x, const float* __restrict__ emb,
                               const float* __restrict__ W_ih, const float* __restrict__ b_ih,
                               const float* __restrict__ b_hh, float* __restrict__ xg) {
    const int nbase = blockIdx.x * 16;
    const int mbase = blockIdx.y * 16;
    const int lane = threadIdx.x;
    const int half = lane >> 4, lr = lane & 15;

    const int m = mbase + lr;                 // this lane's A row
    const int s = m >> 4, b = m & 15;
    const int token = x[b * SEQ + s];
    const float* __restrict__ arow = emb + (long)token * HID;
    const float* __restrict__ brow = W_ih + (long)(nbase + lr) * HID;   // B[k][n] = W_ih[n][k]

    v8f acc = {};
    #pragma unroll 4
    for (int k = 0; k < HID; k += 4) {
        int ks = k + 2 * half;
        float2 av = *(const float2*)(arow + ks);
        float2 bv = *(const float2*)(brow + ks);
        v2f a = {av.x, av.y};
        v2f bb = {bv.x, bv.y};
        acc = wmma_f32(a, bb, acc);
    }
    const int n = nbase + lr;
    const float bias = b_ih[n] + b_hh[n];
    #pragma unroll
    for (int r = 0; r < 8; ++r) {
        int mm = mbase + r + 8 * half;        // row (s*16+b)
        xg[(long)mm * G4 + n] = acc[r] + bias;
    }
}

// ---------------------------------------------------------------------------
// One LSTM time step. Grid: 32 blocks (16 hidden units each), 512 threads
// = 16 waves: wave w -> gate (w>>2), K-chunk (w&3, 128 wide).
// Deterministic LDS reduction, then gate nonlinearity + state update.
__global__ void lstm_step_kernel(const float* __restrict__ xg, const float* __restrict__ W_hh,
                                 const float* __restrict__ h_in, const float* __restrict__ c_in,
                                 float* __restrict__ h_out, float* __restrict__ c_out,
                                 float* __restrict__ hs, int s) {
    __shared__ float pbuf[16 * 256];          // [wave][b*16 + j]
    const int tid = threadIdx.x;
    const int wave = tid >> 5;
    const int lane = tid & 31;
    const int half = lane >> 4, lr = lane & 15;
    const int gate = wave >> 2;
    const int kc = wave & 3;
    const int jbase = blockIdx.x * 16;

    const float* __restrict__ arow = h_in + lr * HID;                                // row b=lr
    const float* __restrict__ brow = W_hh + (long)(gate * HID + jbase + lr) * HID;   // B[k][n]

    v8f acc = {};
    const int k0 = kc * 128;
    #pragma unroll 4
    for (int k = k0; k < k0 + 128; k += 4) {
        int ks = k + 2 * half;
        float2 av = *(const float2*)(arow + ks);
        float2 bv = *(const float2*)(brow + ks);
        v2f a = {av.x, av.y};
        v2f bb = {bv.x, bv.y};
        acc = wmma_f32(a, bb, acc);
    }
    #pragma unroll
    for (int r = 0; r < 8; ++r) {
        int b = r + 8 * half;
        pbuf[wave * 256 + b * 16 + lr] = acc[r];
    }
    __syncthreads();

    if (tid < 256) {
        const int b = tid >> 4, j = tid & 15;
        const long xbase = (long)(s * BATCH + b) * G4 + jbase + j;
        float g0 = xg[xbase + 0 * HID];
        float g1 = xg[xbase + 1 * HID];
        float g2 = xg[xbase + 2 * HID];
        float g3 = xg[xbase + 3 * HID];
        #pragma unroll
        for (int c = 0; c < 4; ++c) {
            g0 += pbuf[(0 * 4 + c) * 256 + tid];
            g1 += pbuf[(1 * 4 + c) * 256 + tid];
            g2 += pbuf[(2 * 4 + c) * 256 + tid];
            g3 += pbuf[(3 * 4 + c) * 256 + tid];
        }
        const int idx = b * HID + jbase + j;
        const float c_old = c_in[idx];
        const float cn = sigm(g1) * c_old + sigm(g0) * tanhf(g2);
        const float hn = sigm(g3) * tanhf(cn);
        c_out[idx] = cn;
        h_out[idx] = hn;
        hs[(long)s * (BATCH * HID) + idx] = hn;   // time-major (S,B,H)
    }
}

// ---------------------------------------------------------------------------
// LayerNorm over H, writing rows in (b*S + s) order (= logits row order).
__global__ void layernorm_kernel(const float* __restrict__ hs,
                                 const float* __restrict__ gamma, const float* __restrict__ beta,
                                 float* __restrict__ out) {
    __shared__ float ssum[256], ssq[256];
    const int r = blockIdx.x;                 // r = b*128 + s
    const int b = r >> 7, s = r & 127;
    const float* __restrict__ row = hs + (long)(s * BATCH + b) * HID;
    const int t = threadIdx.x;
    const float v0 = row[t], v1 = row[t + 256];
    ssum[t] = v0 + v1;
    ssq[t] = v0 * v0 + v1 * v1;
    __syncthreads();
    for (int off = 128; off > 0; off >>= 1) {
        if (t < off) { ssum[t] += ssum[t + off]; ssq[t] += ssq[t + off]; }
        __syncthreads();
    }
    const float mu = ssum[0] * (1.0f / HID);
    const float var = ssq[0] * (1.0f / HID) - mu * mu;
    const float inv = rsqrtf(var + 1e-5f);
    out[(long)r * HID + t]       = (v0 - mu) * inv * gamma[t]       + beta[t];
    out[(long)r * HID + t + 256] = (v1 - mu) * inv * gamma[t + 256] + beta[t + 256];
}

// ---------------------------------------------------------------------------
// logits = out_norm(2048x512) @ fc_W^T(512x32000) + fc_b.
// Each wave: 64x16 tile (4 M-tiles share one B fragment -> 4x fc_W reuse).
__global__ void logits_gemm_kernel(const float* __restrict__ A, const float* __restrict__ fc_W,
                                   const float* __restrict__ fc_b, float* __restrict__ out) {
    const int nbase = blockIdx.x * 16;
    const int mbase = blockIdx.y * 64;
    const int lane = threadIdx.x;
    const int half = lane >> 4, lr = lane & 15;

    const float* __restrict__ brow = fc_W + (long)(nbase + lr) * HID;
    const float* __restrict__ a0 = A + (long)(mbase + 0  + lr) * HID;
    const float* __restrict__ a1 = A + (long)(mbase + 16 + lr) * HID;
    const float* __restrict__ a2 = A + (long)(mbase + 32 + lr) * HID;
    const float* __restrict__ a3 = A + (long)(mbase + 48 + lr) * HID;

    v8f acc0 = {}, acc1 = {}, acc2 = {}, acc3 = {};
    #pragma unroll 2
    for (int k = 0; k < HID; k += 4) {
        int ks = k + 2 * half;
        float2 bv = *(const float2*)(brow + ks);
        v2f bb = {bv.x, bv.y};
        float2 v;
        v = *(const float2*)(a0 + ks); acc0 = wmma_f32((v2f){v.x, v.y}, bb, acc0);
        v = *(const float2*)(a1 + ks); acc1 = wmma_f32((v2f){v.x, v.y}, bb, acc1);
        v = *(const float2*)(a2 + ks); acc2 = wmma_f32((v2f){v.x, v.y}, bb, acc2);
        v = *(const float2*)(a3 + ks); acc3 = wmma_f32((v2f){v.x, v.y}, bb, acc3);
    }
    const int n = nbase + lr;
    const float bias = fc_b[n];
    #pragma unroll
    for (int r = 0; r < 8; ++r) {
        int mm = r + 8 * half;
        out[(long)(mbase + 0  + mm) * VOCAB + n] = acc0[r] + bias;
        out[(long)(mbase + 16 + mm) * VOCAB + n] = acc1[r] + bias;
        out[(long)(mbase + 32 + mm) * VOCAB + n] = acc2[r] + bias;
        out[(long)(mbase + 48 + mm) * VOCAB + n] = acc3[r] + bias;
    }
}

// ---------------------------------------------------------------------------
// hidden_cell tail: (1,B,2H) = [h | c]
__global__ void write_hc_kernel(const float* __restrict__ h, const float* __restrict__ c,
                                float* __restrict__ out_tail) {
    int t = blockIdx.x * 256 + threadIdx.x;   // 0 .. B*2H-1
    if (t >= BATCH * 2 * HID) return;
    int b = t >> 10, j = t & 1023;
    out_tail[t] = (j < HID) ? h[b * HID + j] : c[b * HID + (j - HID)];
}

// ---------------------------------------------------------------------------
extern "C" void kernel_launch(void* const* d_in, const int* in_sizes, int n_in,
                              void* d_out, int out_size, void* d_ws, size_t ws_size,
                              hipStream_t stream) {
    const int*   x     = (const int*)  d_in[0];
    const float* z     = (const float*)d_in[1];
    const float* emb   = (const float*)d_in[2];
    const float* W_ih  = (const float*)d_in[3];
    const float* W_hh  = (const float*)d_in[4];
    const float* b_ih  = (const float*)d_in[5];
    const float* b_hh  = (const float*)d_in[6];
    const float* gamma = (const float*)d_in[7];
    const float* beta  = (const float*)d_in[8];
    const float* fc_W  = (const float*)d_in[9];
    const float* fc_b  = (const float*)d_in[10];
    float* out = (float*)d_out;

    // workspace layout (floats)
    float* ws    = (float*)d_ws;
    float* xg    = ws;                              // S*B*4H = 4,194,304
    float* hs    = xg + (long)SEQ * BATCH * G4;     // S*B*H  = 1,048,576
    float* onorm = hs + (long)SEQ * BATCH * HID;    // B*S*H  = 1,048,576
    float* hb0   = onorm + (long)BATCH * SEQ * HID; // B*H
    float* hb1   = hb0 + BATCH * HID;
    float* cb0   = hb1 + BATCH * HID;
    float* cb1   = cb0 + BATCH * HID;

    init_hc_kernel<<<(BATCH * HID + 255) / 256, 256, 0, stream>>>(z, hb0, cb0);

    xg_gemm_kernel<<<dim3(G4 / 16, (SEQ * BATCH) / 16), 32, 0, stream>>>(
        x, emb, W_ih, b_ih, b_hh, xg);

    for (int s = 0; s < SEQ; ++s) {
        const float* hin  = (s & 1) ? hb1 : hb0;
        const float* cin  = (s & 1) ? cb1 : cb0;
        float*       hout = (s & 1) ? hb0 : hb1;
        float*       cout = (s & 1) ? cb0 : cb1;
        lstm_step_kernel<<<HID / 16, 512, 0, stream>>>(xg, W_hh, hin, cin, hout, cout, hs, s);
    }
    // S=128 even -> final state lands in hb0/cb0

    layernorm_kernel<<<BATCH * SEQ, 256, 0, stream>>>(hs, gamma, beta, onorm);

    logits_gemm_kernel<<<dim3(VOCAB / 16, (BATCH * SEQ) / 64), 32, 0, stream>>>(
        onorm, fc_W, fc_b, out);

    write_hc_kernel<<<(BATCH * 2 * HID + 255) / 256, 256, 0, stream>>>(
        hb0, cb0, out + (long)BATCH * SEQ * VOCAB);
}